// Factorized_Normal_Net_12953621364851
// MI455X (gfx1250) — compile-verified
//
#include <hip/hip_runtime.h>
#include <math.h>

// ---------------------------------------------------------------------------
// Factorized normal net, f32 WMMA path for gfx1250 (MI455X).
//   phi : (phi_embed[8] ++ z[:12]) -> 16 -> 16 -> 1   (Lipschitz-scaled W)
//   th  : (x[:32]      ++ z[:12]) -> 16 -> 16 -> 1
//   out = [sin(th)cos(phi), sin(th)sin(phi), cos(th)] @ tbn   (per point)
// 16 points per wave tile; GEMMs via v_wmma_f32_16x16x4_f32.
// Inputs staged row-major in LDS with padded strides (12/20/36 floats):
// strides are =0 mod 4 with gcd(stride,64)==4, so per-lane column gathers
// S[col*stride + k] are bank-conflict-free across the full wave.
// Staging uses GLOBAL_LOAD_ASYNC_TO_LDS_B128 (ASYNCcnt) when available.
// ---------------------------------------------------------------------------

typedef float v2f __attribute__((ext_vector_type(2)));
typedef float v8f __attribute__((ext_vector_type(8)));
typedef int v4i __attribute__((ext_vector_type(4)));

__device__ __forceinline__ v8f wmma4(v2f a, v2f b, v8f c) {
  // 8-arg form: (neg_a, A, neg_b, B, c_mod, C, reuse_a, reuse_b)
  return __builtin_amdgcn_wmma_f32_16x16x4_f32(false, a, false, b, (short)0, c,
                                               false, false);
}

#if __has_builtin(__builtin_amdgcn_global_load_async_to_lds_b128) && \
    __has_builtin(__builtin_amdgcn_s_wait_asynccnt)
#define FNN_ASYNC_LDS 1
#endif

#ifdef FNN_ASYNC_LDS
// Builtin signature (from hipcc sema): (v4i AS1*, v4i AS3*, imm offset, imm cpol)
typedef __attribute__((address_space(1))) v4i* fnn_gp128;
typedef __attribute__((address_space(3))) v4i* fnn_lp128;
__device__ __forceinline__ void async_cp16(const float* gsrc, float* ldst) {
  __builtin_amdgcn_global_load_async_to_lds_b128((fnn_gp128)gsrc,
                                                 (fnn_lp128)ldst, 0, 0);
}
#endif

// ---------------------------- weight prep ----------------------------------
__device__ __forceinline__ float softplusf(float c) {
  return (c > 20.0f) ? c : log1pf(__expf(c));
}

__device__ void scale_row(const float* W, const float* b, float sp, int row,
                          int cols, float* Wout, float* bout) {
  float s = 0.0f;
  for (int j = 0; j < cols; ++j) s += fabsf(W[row * cols + j]);
  float sc = fminf(1.0f, sp / s);
  for (int j = 0; j < cols; ++j) Wout[row * cols + j] = W[row * cols + j] * sc;
  bout[row] = b[row];
}

// ws layout (floats):
//  0    W1phi[16x20]   320 b1phi[16]   336 W2phi[16x16]  592 b2phi[16]
//  608  w3phi[16]      624 b3phi[1]
//  640  W1th [16x44]  1344 b1th[16]   1360 W2th[16x16]  1616 b2th[16]
//  1632 w3th[16]      1648 b3th[1]
__global__ void fnn_prep(const float* W1p, const float* b1p, const float* c1,
                         const float* W2p, const float* b2p, const float* c2,
                         const float* W3p, const float* b3p, const float* c3,
                         const float* W1t, const float* b1t, const float* c4,
                         const float* W2t, const float* b2t, const float* c5,
                         const float* W3t, const float* b3t, const float* c6,
                         float* ws) {
  int t = threadIdx.x;
  if (t < 16)                  scale_row(W1p, b1p, softplusf(c1[0]), t,      20, ws + 0,    ws + 320);
  else if (t < 32)             scale_row(W2p, b2p, softplusf(c2[0]), t - 16, 16, ws + 336,  ws + 592);
  else if (t == 32)            scale_row(W3p, b3p, softplusf(c3[0]), 0,      16, ws + 608,  ws + 624);
  else if (t >= 64 && t < 80)  scale_row(W1t, b1t, softplusf(c4[0]), t - 64, 44, ws + 640,  ws + 1344);
  else if (t >= 80 && t < 96)  scale_row(W2t, b2t, softplusf(c5[0]), t - 80, 16, ws + 1360, ws + 1616);
  else if (t == 96)            scale_row(W3t, b3t, softplusf(c6[0]), 0,      16, ws + 1632, ws + 1648);
}

// ----------------------------- main kernel ---------------------------------
#define SPH 12  // phi_embed row stride (8 feats + pad), 48B, 16B-aligned
#define SZ 20   // z row stride (16 feats + pad), 80B
#define SX 36   // x row stride (32 feats + pad), 144B
#define LP 17   // hidden bounce stride

__global__ __launch_bounds__(256) void fnn_main(
    const float* __restrict__ z, const float* __restrict__ x,
    const float* __restrict__ phe, const float* __restrict__ tbn,
    const float* __restrict__ wsw, float* __restrict__ out, int npts) {
  __shared__ float sPH[8][16 * SPH];
  __shared__ float sZZ[8][16 * SZ];
  __shared__ float sXX[8][16 * SX];
  __shared__ float sH[8][16 * LP];

  const int lane = threadIdx.x & 31;
  const int wave = threadIdx.x >> 5;
  float* PH = sPH[wave];
  float* ZZ = sZZ[wave];
  float* XX = sXX[wave];
  float* SH = sH[wave];

  const int m = lane & 15;    // A-matrix row / D column
  const int hi = lane >> 4;   // half-wave select
  const int col = lane & 15;  // point within tile

  const float* W1p = wsw + 0;
  const float* b1p = wsw + 320;
  const float* W2p = wsw + 336;
  const float* b2p = wsw + 592;
  const float* w3p = wsw + 608;
  const float* b3p = wsw + 624;
  const float* W1t = wsw + 640;
  const float* b1t = wsw + 1344;
  const float* W2t = wsw + 1360;
  const float* b2t = wsw + 1616;
  const float* w3t = wsw + 1632;
  const float* b3t = wsw + 1648;

  // ---- weight fragments, loaded once per wave (A layout: lanes 0-15 hold
  // K=k0,k0+1; lanes 16-31 hold K=k0+2,k0+3 of row m) ----
  v2f a1p[5], a2p[4], a1t[11], a2t[4];
#pragma unroll
  for (int c = 0; c < 5; ++c) {
    a1p[c].x = W1p[m * 20 + 4 * c + 2 * hi];
    a1p[c].y = W1p[m * 20 + 4 * c + 2 * hi + 1];
  }
#pragma unroll
  for (int c = 0; c < 4; ++c) {
    a2p[c].x = W2p[m * 16 + 4 * c + 2 * hi];
    a2p[c].y = W2p[m * 16 + 4 * c + 2 * hi + 1];
  }
#pragma unroll
  for (int c = 0; c < 11; ++c) {
    a1t[c].x = W1t[m * 44 + 4 * c + 2 * hi];
    a1t[c].y = W1t[m * 44 + 4 * c + 2 * hi + 1];
  }
#pragma unroll
  for (int c = 0; c < 4; ++c) {
    a2t[c].x = W2t[m * 16 + 4 * c + 2 * hi];
    a2t[c].y = W2t[m * 16 + 4 * c + 2 * hi + 1];
  }
  float b1pf[8], b2pf[8], w3pf[8], b1tf[8], b2tf[8], w3tf[8];
#pragma unroll
  for (int v = 0; v < 8; ++v) {
    b1pf[v] = b1p[v + 8 * hi];
    b2pf[v] = b2p[v + 8 * hi];
    w3pf[v] = w3p[v + 8 * hi];
    b1tf[v] = b1t[v + 8 * hi];
    b2tf[v] = b2t[v + 8 * hi];
    w3tf[v] = w3t[v + 8 * hi];
  }
  const float bias3p = b3p[0];
  const float bias3t = b3t[0];

  const int nt = npts >> 4;
  const int gw = (blockIdx.x << 3) + wave;
  const int stride = gridDim.x << 3;

  for (int t = gw; t < nt; t += stride) {
    const int p0 = t << 4;

    // prefetch next tile's inputs into cache hierarchy
    const int tn = t + stride;
    if (tn < nt) {
      const int pn = tn << 4;
      __builtin_prefetch(z + (size_t)pn * 16, 0, 3);
      __builtin_prefetch(x + (size_t)pn * 32, 0, 3);
      __builtin_prefetch(phe + (size_t)pn * 8, 0, 3);
      __builtin_prefetch(tbn + (size_t)pn * 9, 0, 3);
    }

    // ---- stage inputs row-major into LDS (padded strides) ----
    // phi_embed: 128 dwords -> 1x 512B transfer
    // z        : 256 dwords -> 2x        (feats 12..15 staged but unused)
    // x        : 512 dwords -> 4x
#ifdef FNN_ASYNC_LDS
    {
      int g = lane * 4;
      async_cp16(phe + (size_t)p0 * 8 + g, PH + (g >> 3) * SPH + (g & 7));
#pragma unroll
      for (int r = 0; r < 2; ++r) {
        g = r * 128 + lane * 4;
        async_cp16(z + (size_t)p0 * 16 + g, ZZ + (g >> 4) * SZ + (g & 15));
      }
#pragma unroll
      for (int r = 0; r < 4; ++r) {
        g = r * 128 + lane * 4;
        async_cp16(x + (size_t)p0 * 32 + g, XX + (g >> 5) * SX + (g & 31));
      }
    }
    __builtin_amdgcn_s_wait_asynccnt(0);
    __builtin_amdgcn_wave_barrier();
#else
    {
      int g = lane * 4;
      *(float4*)(PH + (g >> 3) * SPH + (g & 7)) =
          *(const float4*)(phe + (size_t)p0 * 8 + g);
#pragma unroll
      for (int r = 0; r < 2; ++r) {
        g = r * 128 + lane * 4;
        *(float4*)(ZZ + (g >> 4) * SZ + (g & 15)) =
            *(const float4*)(z + (size_t)p0 * 16 + g);
      }
#pragma unroll
      for (int r = 0; r < 4; ++r) {
        g = r * 128 + lane * 4;
        *(float4*)(XX + (g >> 5) * SX + (g & 31)) =
            *(const float4*)(x + (size_t)p0 * 32 + g);
      }
    }
    __builtin_amdgcn_wave_barrier();  // same-wave DS ops are in-order
#endif

    // ---------------- phi network ----------------
    // Layer-1 input feature k of point col:
    //   k in [0,8)  -> PH[col*SPH + k]
    //   k in [8,20) -> ZZ[col*SZ + (k-8)]
    v8f d = {0, 0, 0, 0, 0, 0, 0, 0};
#pragma unroll
    for (int c = 0; c < 5; ++c) {  // layer 1, K=20 (chunk source uniform in c)
      v2f b;
      if (4 * c < 8) {
        b.x = PH[col * SPH + 4 * c + 2 * hi];
        b.y = PH[col * SPH + 4 * c + 2 * hi + 1];
      } else {
        b.x = ZZ[col * SZ + 4 * c - 8 + 2 * hi];
        b.y = ZZ[col * SZ + 4 * c - 8 + 2 * hi + 1];
      }
      d = wmma4(a1p[c], b, d);
    }
#pragma unroll
    for (int v = 0; v < 8; ++v) {  // bias + relu -> LDS for D->B reshape
      float h = d[v] + b1pf[v];
      SH[(v + 8 * hi) * LP + col] = h > 0.0f ? h : 0.0f;
    }
    __builtin_amdgcn_wave_barrier();
    v8f d2 = {0, 0, 0, 0, 0, 0, 0, 0};
#pragma unroll
    for (int c = 0; c < 4; ++c) {  // layer 2, K=16
      v2f b;
      b.x = SH[(4 * c + 2 * hi) * LP + col];
      b.y = SH[(4 * c + 2 * hi + 1) * LP + col];
      d2 = wmma4(a2p[c], b, d2);
    }
    float part = 0.0f;
#pragma unroll
    for (int v = 0; v < 8; ++v) {  // layer 3: 16 -> 1 dot
      float h = d2[v] + b2pf[v];
      h = h > 0.0f ? h : 0.0f;
      part = fmaf(w3pf[v], h, part);
    }
    const float phiv = part + __shfl_xor(part, 16, 32) + bias3p;
    __builtin_amdgcn_wave_barrier();

    // ---------------- theta network ----------------
    // Layer-1 input feature k: k in [0,32) -> XX, k in [32,44) -> ZZ
    v8f e1 = {0, 0, 0, 0, 0, 0, 0, 0};
#pragma unroll
    for (int c = 0; c < 11; ++c) {  // layer 1, K=44
      v2f b;
      if (4 * c < 32) {
        b.x = XX[col * SX + 4 * c + 2 * hi];
        b.y = XX[col * SX + 4 * c + 2 * hi + 1];
      } else {
        b.x = ZZ[col * SZ + 4 * c - 32 + 2 * hi];
        b.y = ZZ[col * SZ + 4 * c - 32 + 2 * hi + 1];
      }
      e1 = wmma4(a1t[c], b, e1);
    }
#pragma unroll
    for (int v = 0; v < 8; ++v) {
      float h = e1[v] + b1tf[v];
      SH[(v + 8 * hi) * LP + col] = h > 0.0f ? h : 0.0f;
    }
    __builtin_amdgcn_wave_barrier();
    v8f e2 = {0, 0, 0, 0, 0, 0, 0, 0};
#pragma unroll
    for (int c = 0; c < 4; ++c) {
      v2f b;
      b.x = SH[(4 * c + 2 * hi) * LP + col];
      b.y = SH[(4 * c + 2 * hi + 1) * LP + col];
      e2 = wmma4(a2t[c], b, e2);
    }
    float partT = 0.0f;
#pragma unroll
    for (int v = 0; v < 8; ++v) {
      float h = e2[v] + b2tf[v];
      h = h > 0.0f ? h : 0.0f;
      partT = fmaf(w3tf[v], h, partT);
    }
    const float thv = partT + __shfl_xor(partT, 16, 32) + bias3t;
    __builtin_amdgcn_wave_barrier();

    // ---------------- normal + TBN rotate ----------------
    const float st = __sinf(thv), ct = __cosf(thv);
    const float sp = __sinf(phiv), cp = __cosf(phiv);
    const float nx = st * cp, ny = st * sp, nz = ct;
    if (lane < 16) {
      const float* tb = tbn + (size_t)(p0 + col) * 9;
      float* op = out + (size_t)(p0 + col) * 3;
      op[0] = nx * tb[0] + ny * tb[3] + nz * tb[6];
      op[1] = nx * tb[1] + ny * tb[4] + nz * tb[7];
      op[2] = nx * tb[2] + ny * tb[5] + nz * tb[8];
    }
  }
}

// ----------------------------- launcher ------------------------------------
extern "C" void kernel_launch(void* const* d_in, const int* in_sizes, int n_in,
                              void* d_out, int out_size, void* d_ws,
                              size_t ws_size, hipStream_t stream) {
  const float* z = (const float*)d_in[0];
  const float* x = (const float*)d_in[1];
  const float* phe = (const float*)d_in[2];
  const float* tbn = (const float*)d_in[3];
  float* ws = (float*)d_ws;

  fnn_prep<<<1, 128, 0, stream>>>(
      (const float*)d_in[4], (const float*)d_in[5], (const float*)d_in[6],
      (const float*)d_in[7], (const float*)d_in[8], (const float*)d_in[9],
      (const float*)d_in[10], (const float*)d_in[11], (const float*)d_in[12],
      (const float*)d_in[13], (const float*)d_in[14], (const float*)d_in[15],
      (const float*)d_in[16], (const float*)d_in[17], (const float*)d_in[18],
      (const float*)d_in[19], (const float*)d_in[20], (const float*)d_in[21],
      ws);

  const int npts = in_sizes[2] / 8;  // phi_embed is (N, 8)
  const int nt = npts / 16;
  int blocks = (nt + 7) / 8;  // 8 waves per block
  if (blocks > 1024) blocks = 1024;
  if (blocks < 1) blocks = 1;
  fnn_main<<<blocks, 256, 0, stream>>>(z, x, phe, tbn, ws, (float*)d_out, npts);
}